// AttentionAddition_19129784336475
// MI455X (gfx1250) — compile-verified
//
#include <hip/hip_runtime.h>
#include <hip/hip_bf16.h>

#define B_ 16
#define T_ 4096
#define E_ 768
#define NEGINF (-10000000000.0f)
#define NCHUNK 32
#define CHUNK_T (T_ / NCHUNK)   // 128

typedef __attribute__((ext_vector_type(2))) float v2f;
typedef __attribute__((ext_vector_type(4))) float v4f;
typedef __attribute__((ext_vector_type(8))) float v8f;

// ---------------------------------------------------------------------------
// Kernel 1: qscore[b] = sum_e tanh(q[b,e]) * v_w[e]       (16 blocks, trivial)
// ---------------------------------------------------------------------------
__global__ __launch_bounds__(256) void qscore_kernel(
    const float* __restrict__ q, const float* __restrict__ v_w,
    float* __restrict__ qscore) {
  __shared__ float red[256];
  const int b = blockIdx.x;
  const int tid = threadIdx.x;
  float acc = 0.0f;
  for (int e = tid; e < E_; e += 256)
    acc += tanhf(q[b * E_ + e]) * v_w[e];
  red[tid] = acc;
  __syncthreads();
  for (int s = 128; s > 0; s >>= 1) {
    if (tid < s) red[tid] += red[tid + s];
    __syncthreads();
  }
  if (tid == 0) qscore[b] = red[0];
}

// ---------------------------------------------------------------------------
// Kernel 2: scores[b,t] = qscore[b] + tanh(c[b,t,:]) . v_w[E:], masked.
// One wave32 per t-row; float4 coalesced streaming of context (201 MB).
// ---------------------------------------------------------------------------
__global__ __launch_bounds__(256) void score_kernel(
    const float* __restrict__ ctx, const float* __restrict__ v_w,
    const int* __restrict__ mask, const float* __restrict__ qscore,
    float* __restrict__ scores) {
  const int idx = blockIdx.x;              // b * (T/8) + group
  const int b = idx / (T_ / 8);
  const int grp = idx % (T_ / 8);
  const int wave = threadIdx.x >> 5;
  const int lane = threadIdx.x & 31;
  const int t = grp * 8 + wave;

  const float* row = ctx + ((size_t)b * T_ + t) * E_;
  const float* w2 = v_w + E_;

  float acc = 0.0f;
#pragma unroll
  for (int i = 0; i < E_ / 128; ++i) {     // 6 iterations: 32 lanes * float4
    const int e0 = i * 128 + lane * 4;
    v4f c = *(const v4f*)(row + e0);
    v4f w = *(const v4f*)(w2 + e0);
    acc += tanhf(c[0]) * w[0] + tanhf(c[1]) * w[1] +
           tanhf(c[2]) * w[2] + tanhf(c[3]) * w[3];
  }
  // wave32 butterfly reduction
  for (int off = 16; off > 0; off >>= 1) acc += __shfl_xor(acc, off, 32);

  if (lane == 0) {
    float s = qscore[b] + acc;
    if (mask[(size_t)b * T_ + t] < 1) s = NEGINF;
    scores[(size_t)b * T_ + t] = s;
  }
}

// ---------------------------------------------------------------------------
// Kernel 3: softmax over t per batch. 16 blocks x 1024 threads, 4 t each.
// ---------------------------------------------------------------------------
__global__ __launch_bounds__(1024) void softmax_kernel(
    const float* __restrict__ scores, float* __restrict__ attn) {
  __shared__ float red[1024];
  const int b = blockIdx.x;
  const int tid = threadIdx.x;

  float v[4];
#pragma unroll
  for (int i = 0; i < 4; ++i) v[i] = scores[(size_t)b * T_ + tid + i * 1024];

  float m = fmaxf(fmaxf(v[0], v[1]), fmaxf(v[2], v[3]));
  red[tid] = m;
  __syncthreads();
  for (int s = 512; s > 0; s >>= 1) {
    if (tid < s) red[tid] = fmaxf(red[tid], red[tid + s]);
    __syncthreads();
  }
  m = red[0];
  __syncthreads();

  float e[4];
  float lsum = 0.0f;
#pragma unroll
  for (int i = 0; i < 4; ++i) {
    e[i] = __expf(v[i] - m);
    lsum += e[i];
  }
  red[tid] = lsum;
  __syncthreads();
  for (int s = 512; s > 0; s >>= 1) {
    if (tid < s) red[tid] += red[tid + s];
    __syncthreads();
  }
  const float inv = 1.0f / red[0];
#pragma unroll
  for (int i = 0; i < 4; ++i)
    attn[(size_t)b * T_ + tid + i * 1024] = e[i] * inv;
}

// ---------------------------------------------------------------------------
// Kernel 4: partial[b,chunk,e] = sum_{t in chunk} attn[b,t] * c[b,t,e]
// via V_WMMA_F32_16X16X4_F32. Per WMMA: A row 0 = attn[kk..kk+3] (rest 0),
// B = 4x16 context tile. D row 0 (acc VGPR0, lanes 0-15) is the result row.
// 512 blocks x 8 waves; each wave owns 96 e-columns (6 tiles of 16).
// ---------------------------------------------------------------------------
__global__ __launch_bounds__(256) void wsum_wmma_kernel(
    const float* __restrict__ ctx, const float* __restrict__ attn,
    float* __restrict__ partials) {
  __shared__ float s_attn[CHUNK_T];
  const int b = blockIdx.x / NCHUNK;
  const int chunk = blockIdx.x % NCHUNK;
  const int t0 = chunk * CHUNK_T;
  const int tid = threadIdx.x;

  if (tid < CHUNK_T) s_attn[tid] = attn[(size_t)b * T_ + t0 + tid];
  __syncthreads();

  const int wave = tid >> 5;
  const int lane = tid & 31;
  const int halfsel = lane >> 4;   // 0: lanes 0-15 (K=0,1) ; 1: lanes 16-31 (K=2,3)
  const int nlane = lane & 15;     // N / M index within half-wave
  const int ebase = wave * 96;     // 6 e-tiles of 16 per wave

  const float* cb = ctx + ((size_t)b * T_ + t0) * E_;

  v8f acc[6];
#pragma unroll
  for (int j = 0; j < 6; ++j) acc[j] = (v8f)(0.0f);

  for (int kk = 0; kk < CHUNK_T; kk += 4) {
    const int ra = kk + (halfsel << 1);
    // A-matrix: only M=0 row non-zero -> lanes with nlane==0 carry attn.
    const float av0 = s_attn[ra];
    const float av1 = s_attn[ra + 1];
    v2f a;
    a[0] = (nlane == 0) ? av0 : 0.0f;   // K = 0 (lanes 0-15) / K = 2 (lanes 16-31)
    a[1] = (nlane == 0) ? av1 : 0.0f;   // K = 1 / K = 3

    const float* rp = cb + (size_t)ra * E_ + ebase + nlane;
#pragma unroll
    for (int j = 0; j < 6; ++j) {
      v2f bv;
      bv[0] = rp[j * 16];        // B VGPR0: rows K=0 / K=2
      bv[1] = rp[j * 16 + E_];   // B VGPR1: rows K=1 / K=3
      acc[j] = __builtin_amdgcn_wmma_f32_16x16x4_f32(
          false, a, false, bv, (short)0, acc[j], false, false);
    }
  }

  // D row M=0 lives in acc[j][0] on lanes 0..15 (N = lane).
  if (lane < 16) {
    float* pout = partials + ((size_t)b * NCHUNK + chunk) * E_ + ebase + nlane;
#pragma unroll
    for (int j = 0; j < 6; ++j) pout[j * 16] = acc[j][0];
  }
}

// ---------------------------------------------------------------------------
// Kernel 5: out[b,e] = sum_chunk partials[b,chunk,e]   (48 blocks, trivial)
// ---------------------------------------------------------------------------
__global__ __launch_bounds__(256) void reduce_kernel(
    const float* __restrict__ partials, float* __restrict__ out) {
  const int idx = blockIdx.x * 256 + threadIdx.x;  // < B*E
  const int b = idx / E_;
  const int e = idx % E_;
  float s = 0.0f;
#pragma unroll 8
  for (int ch = 0; ch < NCHUNK; ++ch)
    s += partials[((size_t)b * NCHUNK + ch) * E_ + e];
  out[idx] = s;
}

// ---------------------------------------------------------------------------
extern "C" void kernel_launch(void* const* d_in, const int* in_sizes, int n_in,
                              void* d_out, int out_size, void* d_ws,
                              size_t ws_size, hipStream_t stream) {
  const float* query = (const float*)d_in[0];    // [B,E]
  const float* context = (const float*)d_in[1];  // [B,T,E]
  const int* mask = (const int*)d_in[2];         // [B,T]
  const float* v_w = (const float*)d_in[3];      // [2E]
  float* out = (float*)d_out;                    // [B,E]

  // Workspace layout (floats)
  float* ws = (float*)d_ws;
  float* qscore = ws;                 // 16
  float* scores = qscore + 16;        // B*T
  float* attn = scores + B_ * T_;     // B*T
  float* partials = attn + B_ * T_;   // B*NCHUNK*E

  (void)in_sizes; (void)n_in; (void)out_size; (void)ws_size;

  qscore_kernel<<<B_, 256, 0, stream>>>(query, v_w, qscore);
  score_kernel<<<B_ * (T_ / 8), 256, 0, stream>>>(context, v_w, mask, qscore,
                                                  scores);
  softmax_kernel<<<B_, 1024, 0, stream>>>(scores, attn);
  wsum_wmma_kernel<<<B_ * NCHUNK, 256, 0, stream>>>(context, attn, partials);
  reduce_kernel<<<(B_ * E_) / 256, 256, 0, stream>>>(partials, out);
}